// PointTransformerEncoder_2241972928925
// MI455X (gfx1250) — compile-verified
//
#include <hip/hip_runtime.h>
#include <cstddef>
#include <cstdint>

// ---------------------------------------------------------------------------
// PointTransformer encoder for MI455X (gfx1250), fp32 WMMA (V_WMMA_F32_16X16X4_F32)
// ---------------------------------------------------------------------------

typedef float v2f __attribute__((ext_vector_type(2)));
typedef float v8f __attribute__((ext_vector_type(8)));

#define KNN_K   16
#define PT_HID  64
#define NBATCH  16

__device__ __forceinline__ v8f wmma4(v2f a, v2f b, v8f c) {
  // D = A(16x4,f32) * B(4x16,f32) + C(16x16,f32)
  return __builtin_amdgcn_wmma_f32_16x16x4_f32(false, a, false, b, (short)0, c, false, false);
}

// ---------------------------------------------------------------------------
// Generic tiled GEMM: Y[M,N] = act(A[M,K] @ W[K,N] + bias)
// grid = (N/64, M/16), block = 128 (4 waves, one 16x16 n-tile per wave)
// Requires: M %16==0, N %16==0 (launched with N%64==0 here), K %4==0
// ---------------------------------------------------------------------------
__global__ void gemm_wmma_kernel(const float* __restrict__ A, const float* __restrict__ W,
                                 const float* __restrict__ bias, float* __restrict__ Y,
                                 int M, int K, int N, int act) {
  const int wave = threadIdx.x >> 5;
  const int lane = threadIdx.x & 31;
  const int lm   = lane & 15;   // row (A) / col (B,D)
  const int lh   = lane >> 4;   // lane half
  const int m0   = blockIdx.y * 16;
  const int n0   = (blockIdx.x * 4 + wave) * 16;
  if (n0 >= N) return;

  const float* Arow = A + (size_t)(m0 + lm) * K;
  v8f acc = {};
  for (int kk = 0; kk < K; kk += 4) {
    const int k0 = kk + 2 * lh;
    v2f a, b;
    a.x = Arow[k0];
    a.y = Arow[k0 + 1];
    b.x = W[(size_t)k0 * N + n0 + lm];
    b.y = W[(size_t)(k0 + 1) * N + n0 + lm];
    acc = wmma4(a, b, acc);
  }
  const int col = n0 + lm;
  const float bv = bias ? bias[col] : 0.0f;
#pragma unroll
  for (int j = 0; j < 8; ++j) {
    float v = acc[j] + bv;
    if (act) v = fmaxf(v, 0.0f);
    Y[(size_t)(m0 + j + 8 * lh) * N + col] = v;
  }
}

// ---------------------------------------------------------------------------
// Batch-norm statistics: per-channel mean & rsqrt(var+eps) over M rows.
// grid = C, block = 256
// ---------------------------------------------------------------------------
__global__ void bn_stats_kernel(const float* __restrict__ X, float* __restrict__ mean,
                                float* __restrict__ rs, int M, int C) {
  const int c = blockIdx.x;
  float s = 0.0f, sq = 0.0f;
  for (int i = threadIdx.x; i < M; i += blockDim.x) {
    float v = X[(size_t)i * C + c];
    s += v; sq += v * v;
  }
  __shared__ float sh_s[256];
  __shared__ float sh_q[256];
  sh_s[threadIdx.x] = s; sh_q[threadIdx.x] = sq;
  __syncthreads();
  for (int st = 128; st > 0; st >>= 1) {
    if ((int)threadIdx.x < st) {
      sh_s[threadIdx.x] += sh_s[threadIdx.x + st];
      sh_q[threadIdx.x] += sh_q[threadIdx.x + st];
    }
    __syncthreads();
  }
  if (threadIdx.x == 0) {
    float mu  = sh_s[0] / (float)M;
    float var = sh_q[0] / (float)M - mu * mu;
    mean[c] = mu;
    rs[c]   = rsqrtf(var + 1e-5f);
  }
}

__global__ void bn_apply_relu_kernel(float* __restrict__ X, const float* __restrict__ mean,
                                     const float* __restrict__ rs, const float* __restrict__ gamma,
                                     const float* __restrict__ beta, size_t total, int C) {
  size_t i = (size_t)blockIdx.x * blockDim.x + threadIdx.x;
  if (i >= total) return;
  int c = (int)(i % C);
  float v = (X[i] - mean[c]) * rs[c] * gamma[c] + beta[c];
  X[i] = fmaxf(v, 0.0f);
}

// ---------------------------------------------------------------------------
// Brute-force kNN (k=16). One thread per query. total = B*Nq.
// Tie-break: smaller base index first (matches lax.top_k on -d).
// ---------------------------------------------------------------------------
__global__ void knn_kernel(const float* __restrict__ q, const float* __restrict__ base,
                           int* __restrict__ out, int Nq, int Nb, int excl, int total) {
  const int gid = blockIdx.x * blockDim.x + threadIdx.x;
  if (gid >= total) return;
  const int b = gid / Nq;
  const int i = gid - b * Nq;
  const float qx = q[(size_t)gid * 3 + 0];
  const float qy = q[(size_t)gid * 3 + 1];
  const float qz = q[(size_t)gid * 3 + 2];
  float dist[KNN_K];
  int   idx[KNN_K];
#pragma unroll
  for (int s = 0; s < KNN_K; ++s) { dist[s] = 3.0e38f; idx[s] = 0; }
  const float* bp = base + (size_t)b * Nb * 3;
  for (int j = 0; j < Nb; ++j) {
    if (excl && j == i) continue;
    float dx = qx - bp[j * 3 + 0];
    float dy = qy - bp[j * 3 + 1];
    float dz = qz - bp[j * 3 + 2];
    float d = dx * dx + dy * dy + dz * dz;
    if (d < dist[KNN_K - 1]) {
      dist[KNN_K - 1] = d; idx[KNN_K - 1] = j;
#pragma unroll
      for (int s = KNN_K - 1; s > 0; --s) {
        if (dist[s] < dist[s - 1]) {
          float td = dist[s]; dist[s] = dist[s - 1]; dist[s - 1] = td;
          int   ti = idx[s];  idx[s]  = idx[s - 1];  idx[s - 1]  = ti;
        }
      }
    }
  }
#pragma unroll
  for (int s = 0; s < KNN_K; ++s) out[(size_t)gid * KNN_K + s] = idx[s];
}

// ---------------------------------------------------------------------------
// Farthest point sampling. One block (256 thr) per batch; start at index 0.
// Tie-break: first (smallest) index, matching jnp.argmax.
// ---------------------------------------------------------------------------
__global__ void fps_kernel(const float* __restrict__ pos, int* __restrict__ idx_out,
                           int N, int nsub) {
  __shared__ float s_pos[1024 * 3];
  __shared__ float s_dmin[1024];
  __shared__ float s_rv[256];
  __shared__ int   s_ri[256];
  __shared__ float s_sel[3];
  const int b = blockIdx.x;
  const int t = threadIdx.x;
  const float* P = pos + (size_t)b * N * 3;
  for (int j = t; j < N; j += 256) {
    s_pos[j * 3 + 0] = P[j * 3 + 0];
    s_pos[j * 3 + 1] = P[j * 3 + 1];
    s_pos[j * 3 + 2] = P[j * 3 + 2];
  }
  __syncthreads();
  const float x0 = s_pos[0], y0 = s_pos[1], z0 = s_pos[2];
  for (int j = t; j < N; j += 256) {
    float dx = s_pos[j * 3 + 0] - x0;
    float dy = s_pos[j * 3 + 1] - y0;
    float dz = s_pos[j * 3 + 2] - z0;
    s_dmin[j] = dx * dx + dy * dy + dz * dz;
  }
  if (t == 0) idx_out[(size_t)b * nsub + 0] = 0;
  __syncthreads();
  for (int i = 1; i < nsub; ++i) {
    float bv = -1.0f; int bi = 0x7fffffff;
    for (int j = t; j < N; j += 256) {
      float v = s_dmin[j];
      if (v > bv) { bv = v; bi = j; }
    }
    s_rv[t] = bv; s_ri[t] = bi;
    __syncthreads();
    for (int st = 128; st > 0; st >>= 1) {
      if (t < st) {
        float v2 = s_rv[t + st]; int i2 = s_ri[t + st];
        if (v2 > s_rv[t] || (v2 == s_rv[t] && i2 < s_ri[t])) { s_rv[t] = v2; s_ri[t] = i2; }
      }
      __syncthreads();
    }
    if (t == 0) {
      int sel = s_ri[0];
      idx_out[(size_t)b * nsub + i] = sel;
      s_sel[0] = s_pos[sel * 3 + 0];
      s_sel[1] = s_pos[sel * 3 + 1];
      s_sel[2] = s_pos[sel * 3 + 2];
    }
    __syncthreads();
    const float sx = s_sel[0], sy = s_sel[1], sz = s_sel[2];
    for (int j = t; j < N; j += 256) {
      float dx = s_pos[j * 3 + 0] - sx;
      float dy = s_pos[j * 3 + 1] - sy;
      float dz = s_pos[j * 3 + 2] - sz;
      float d = dx * dx + dy * dy + dz * dz;
      if (d < s_dmin[j]) s_dmin[j] = d;
    }
    __syncthreads();
  }
}

__global__ void gather_pos_kernel(const float* __restrict__ pos, const int* __restrict__ idx,
                                  float* __restrict__ out, int N, int nsub, int total) {
  const int gid = blockIdx.x * blockDim.x + threadIdx.x;  // over B*nsub
  if (gid >= total) return;
  const int b = gid / nsub;
  const int j = idx[gid];
  const float* p = pos + ((size_t)b * N + j) * 3;
  out[(size_t)gid * 3 + 0] = p[0];
  out[(size_t)gid * 3 + 1] = p[1];
  out[(size_t)gid * 3 + 2] = p[2];
}

// max over k gathered neighbors (TransitionDown scatter-max).
__global__ void maxpool_kernel(const float* __restrict__ xm, const int* __restrict__ nbd,
                               float* __restrict__ out, int Np, int nsub, int C, size_t total) {
  size_t gid = (size_t)blockIdx.x * blockDim.x + threadIdx.x;  // over B*nsub*C
  if (gid >= total) return;
  const int c = (int)(gid % C);
  const size_t q = gid / C;           // b*nsub + qq
  const int b = (int)(q / nsub);
  const int* nn = nbd + q * KNN_K;
  float m = -3.0e38f;
#pragma unroll
  for (int e = 0; e < KNN_K; ++e) {
    float v = xm[((size_t)b * Np + nn[e]) * C + c];
    m = fmaxf(m, v);
  }
  out[gid] = m;
}

__global__ void meanpool_kernel(const float* __restrict__ X, float* __restrict__ out,
                                int Np, int C) {
  const int b = blockIdx.x;
  for (int c = threadIdx.x; c < C; c += blockDim.x) {
    float s = 0.0f;
    for (int n = 0; n < Np; ++n) s += X[((size_t)b * Np + n) * C + c];
    out[b * C + c] = s / (float)Np;
  }
}

// ---------------------------------------------------------------------------
// Fused PointTransformerConv: one 128-thread block per target node.
// k=16 edges == WMMA M dimension. pos_nn, alpha, attn_nn, softmax-over-k and
// the weighted sum are fully fused; softmax happens inside the D-tile regs.
// dynamic LDS: 16*C (delta) + 16*64 (hidden) + C (a_dst) + 64 (rel) floats + 16 ints.
// ---------------------------------------------------------------------------
__global__ void ptconv_kernel(const float* __restrict__ pos, const int* __restrict__ nb,
                              const float* __restrict__ v, const float* __restrict__ asrc,
                              const float* __restrict__ adst,
                              const float* __restrict__ pw1, const float* __restrict__ pb1,
                              const float* __restrict__ pw2, const float* __restrict__ pb2,
                              const float* __restrict__ aw1, const float* __restrict__ ab1,
                              const float* __restrict__ aw2, const float* __restrict__ ab2,
                              float* __restrict__ out, int Np, int C) {
  extern __shared__ float smem[];
  float* s_delta = smem;                     // 16*C
  float* s_h     = s_delta + 16 * C;         // 16*64
  float* s_adst  = s_h + 16 * PT_HID;        // C
  float* s_rel   = s_adst + C;               // 16*4
  int*   s_nbi   = (int*)(s_rel + 64);       // 16

  const int node = blockIdx.x;               // b*Np + n
  const int b    = node / Np;
  const int tid  = threadIdx.x;
  const int wave = tid >> 5;
  const int lane = tid & 31;
  const int lm   = lane & 15;
  const int lh   = lane >> 4;

  // stage 0: neighbor ids, relative positions, a_dst row
  if (tid < 16) {
    const int j = nb[(size_t)node * KNN_K + tid];
    s_nbi[tid] = j;
    const float* pn = pos + (size_t)node * 3;
    const float* pj = pos + ((size_t)b * Np + j) * 3;
    s_rel[tid * 4 + 0] = pn[0] - pj[0];
    s_rel[tid * 4 + 1] = pn[1] - pj[1];
    s_rel[tid * 4 + 2] = pn[2] - pj[2];
    s_rel[tid * 4 + 3] = 0.0f;
  }
  for (int c = tid; c < C; c += blockDim.x) s_adst[c] = adst[(size_t)node * C + c];
  __syncthreads();

  // pos_nn layer 1: h[16][64] = relu(rel(16x4pad) @ pw1(3x64) + pb1). 4 n-tiles, 1/wave.
  {
    const int col = wave * 16 + lm;
    const int k0  = 2 * lh;
    v2f a, bb;
    a.x = s_rel[lm * 4 + k0];
    a.y = s_rel[lm * 4 + k0 + 1];
    bb.x = (k0     < 3) ? pw1[(size_t)k0 * PT_HID + col]       : 0.0f;
    bb.y = (k0 + 1 < 3) ? pw1[(size_t)(k0 + 1) * PT_HID + col] : 0.0f;
    v8f acc = {};
    acc = wmma4(a, bb, acc);
#pragma unroll
    for (int j = 0; j < 8; ++j) {
      const int m = j + 8 * lh;
      s_h[m * PT_HID + col] = fmaxf(acc[j] + pb1[col], 0.0f);
    }
  }
  __syncthreads();

  // pos_nn layer 2: delta[16][C] = relu(h(16x64) @ pw2(64xC) + pb2)
  {
    const int ntiles = C >> 4;
    for (int t = wave; t < ntiles; t += 4) {
      const int col = t * 16 + lm;
      v8f acc = {};
      for (int kk = 0; kk < PT_HID; kk += 4) {
        const int k0 = kk + 2 * lh;
        v2f a, bb;
        a.x = s_h[lm * PT_HID + k0];
        a.y = s_h[lm * PT_HID + k0 + 1];
        bb.x = pw2[(size_t)k0 * C + col];
        bb.y = pw2[(size_t)(k0 + 1) * C + col];
        acc = wmma4(a, bb, acc);
      }
#pragma unroll
      for (int j = 0; j < 8; ++j) {
        const int m = j + 8 * lh;
        s_delta[m * C + col] = fmaxf(acc[j] + pb2[col], 0.0f);
      }
    }
  }
  __syncthreads();

  // attn_nn layer 1: h[16][64] = relu(alpha(16xC) @ aw1(Cx64) + ab1),
  // alpha formed on-the-fly: a_dst[n] - a_src[nb_m] + delta[m].
  {
    const int col = wave * 16 + lm;
    const int row = s_nbi[lm];
    const float* as = asrc + ((size_t)b * Np + row) * C;
    v8f acc = {};
    for (int kk = 0; kk < C; kk += 4) {
      const int k0 = kk + 2 * lh;
      v2f a, bb;
      a.x = s_adst[k0]     - as[k0]     + s_delta[lm * C + k0];
      a.y = s_adst[k0 + 1] - as[k0 + 1] + s_delta[lm * C + k0 + 1];
      bb.x = aw1[(size_t)k0 * PT_HID + col];
      bb.y = aw1[(size_t)(k0 + 1) * PT_HID + col];
      acc = wmma4(a, bb, acc);
    }
#pragma unroll
    for (int j = 0; j < 8; ++j) {
      const int m = j + 8 * lh;
      s_h[m * PT_HID + col] = fmaxf(acc[j] + ab1[col], 0.0f);
    }
  }
  __syncthreads();

  // attn_nn layer 2 + softmax over the 16 edges + weighted sum, per 16-channel tile.
  {
    const int ntiles = C >> 4;
    for (int t = wave; t < ntiles; t += 4) {
      const int col = t * 16 + lm;
      v8f acc = {};
      for (int kk = 0; kk < PT_HID; kk += 4) {
        const int k0 = kk + 2 * lh;
        v2f a, bb;
        a.x = s_h[lm * PT_HID + k0];
        a.y = s_h[lm * PT_HID + k0 + 1];
        bb.x = aw2[(size_t)k0 * C + col];
        bb.y = aw2[(size_t)(k0 + 1) * C + col];
        acc = wmma4(a, bb, acc);
      }
      // D-tile: lane holds edges m = j + 8*lh for channel `col`.
      float lg[8];
      float mx = -3.0e38f;
#pragma unroll
      for (int j = 0; j < 8; ++j) {
        lg[j] = fmaxf(acc[j] + ab2[col], 0.0f);  // relu before softmax
        mx = fmaxf(mx, lg[j]);
      }
      mx = fmaxf(mx, __shfl_xor(mx, 16, 32));
      float sum = 0.0f;
#pragma unroll
      for (int j = 0; j < 8; ++j) { lg[j] = __expf(lg[j] - mx); sum += lg[j]; }
      sum += __shfl_xor(sum, 16, 32);
      const float inv = 1.0f / sum;
      float o = 0.0f;
#pragma unroll
      for (int j = 0; j < 8; ++j) {
        const int m = j + 8 * lh;
        const int row = s_nbi[m];
        const float vj = v[((size_t)b * Np + row) * C + col];
        o += lg[j] * inv * (vj + s_delta[m * C + col]);
      }
      o += __shfl_xor(o, 16, 32);
      if (lh == 0) out[(size_t)node * C + col] = o;
    }
  }
}

// ---------------------------------------------------------------------------
// Host orchestration
// ---------------------------------------------------------------------------
struct TBParams {
  const float *w_in, *b_in, *w_out, *b_out, *w_lin, *w_src, *w_dst;
  const float *pw1, *pb1, *pw2, *pb2, *aw1, *ab1, *aw2, *ab2;
};

extern "C" void kernel_launch(void* const* d_in, const int* in_sizes, int n_in,
                              void* d_out, int out_size, void* d_ws, size_t ws_size,
                              hipStream_t stream) {
  (void)out_size; (void)ws_size;

  // --- locate x / pos / params robustly against pytree flattening order ---
  const bool xfirst = (in_sizes[0] == NBATCH * 1024 * 32);  // 524288 unique to x
  const int pbase   = xfirst ? 2 : 0;
  const float* d_x   = (const float*)d_in[xfirst ? 0 : (n_in - 1)];
  const float* d_pos = (const float*)d_in[xfirst ? 1 : (n_in - 2)];
  auto P = [&](int i) { return (const float*)d_in[pbase + i]; };
  // sorted-keys pytree => mlp_in.b (size 128) first; insertion order => mlp_in.w (4096)
  const bool skeys = (in_sizes[pbase] == 128);

  int base_mlp_in, base_t_in, base_tds0, base_tds1, base_tfs0, base_tfs1, base_mlp_out;
  if (skeys) { base_mlp_in = 0; base_mlp_out = 4; base_t_in = 6; base_tds0 = 21; base_tds1 = 25; base_tfs0 = 29; base_tfs1 = 44; }
  else       { base_mlp_in = 0; base_t_in = 4; base_tds0 = 19; base_tds1 = 23; base_tfs0 = 27; base_tfs1 = 42; base_mlp_out = 57; }

  auto bn_w     = [&](int base) { return skeys ? P(base + 3) : P(base + 0); };
  auto bn_b     = [&](int base) { return skeys ? P(base + 0) : P(base + 1); };
  auto bn_gamma = [&](int base) { return P(base + 2); };
  auto bn_beta  = [&](int base) { return skeys ? P(base + 1) : P(base + 3); };
  const float* head_w = skeys ? P(base_mlp_out + 1) : P(base_mlp_out + 0);
  const float* head_b = skeys ? P(base_mlp_out + 0) : P(base_mlp_out + 1);

  auto tbp = [&](int base) {
    TBParams t;
    if (skeys) {
      t.ab1 = P(base + 0);  t.ab2 = P(base + 1);  t.aw1 = P(base + 2);  t.aw2 = P(base + 3);
      t.b_in = P(base + 4); t.b_out = P(base + 5);
      t.pb1 = P(base + 6);  t.pb2 = P(base + 7);  t.pw1 = P(base + 8);  t.pw2 = P(base + 9);
      t.w_dst = P(base + 10); t.w_in = P(base + 11); t.w_lin = P(base + 12);
      t.w_out = P(base + 13); t.w_src = P(base + 14);
    } else {
      t.w_in = P(base + 0); t.b_in = P(base + 1); t.w_out = P(base + 2); t.b_out = P(base + 3);
      t.w_lin = P(base + 4); t.w_src = P(base + 5); t.w_dst = P(base + 6);
      t.pw1 = P(base + 7); t.pb1 = P(base + 8); t.pw2 = P(base + 9); t.pb2 = P(base + 10);
      t.aw1 = P(base + 11); t.ab1 = P(base + 12); t.aw2 = P(base + 13); t.ab2 = P(base + 14);
    }
    return t;
  };

  // --- workspace arena ---
  char* ws = (char*)d_ws;
  size_t off = 0;
  auto alloc = [&](size_t bytes) -> void* {
    void* p = ws + off;
    off += (bytes + 255) & ~(size_t)255;
    return p;
  };
  float* f_x    = (float*)alloc((size_t)8 << 20);    // current features (<= 2M fl)
  float* f_tmp  = (float*)alloc((size_t)16 << 20);   // h / conv / xm
  float* f_v    = (float*)alloc((size_t)8 << 20);
  float* f_asrc = (float*)alloc((size_t)8 << 20);
  float* f_adst = (float*)alloc((size_t)8 << 20);
  float* f_pos0 = (float*)alloc((size_t)NBATCH * 1024 * 3 * 4);
  float* f_pos1 = (float*)alloc((size_t)NBATCH * 1024 * 3 * 4);
  int*   i_nb   = (int*)alloc((size_t)NBATCH * 1024 * KNN_K * 4);
  int*   i_nbd  = (int*)alloc((size_t)NBATCH * 1024 * KNN_K * 4);
  int*   i_fps  = (int*)alloc((size_t)NBATCH * 512 * 4);
  float* f_mean = (float*)alloc(512 * 4);
  float* f_rs   = (float*)alloc(512 * 4);
  float* f_pool = (float*)alloc((size_t)NBATCH * 512 * 4);
  float* f_conv = f_tmp + (((size_t)8 << 20) / sizeof(float));

  auto gemm = [&](const float* A, const float* W, const float* bias, float* Y,
                  int M, int K, int N, int act) {
    dim3 g((N + 63) / 64, M / 16);
    gemm_wmma_kernel<<<g, dim3(128), 0, stream>>>(A, W, bias, Y, M, K, N, act);
  };

  auto bn_mlp = [&](const float* x_in, float* y, int M, int K, int N, int base) {
    gemm(x_in, bn_w(base), bn_b(base), y, M, K, N, 0);
    bn_stats_kernel<<<dim3(N), dim3(256), 0, stream>>>(y, f_mean, f_rs, M, N);
    size_t total = (size_t)M * N;
    bn_apply_relu_kernel<<<dim3((unsigned)((total + 255) / 256)), dim3(256), 0, stream>>>(
        y, f_mean, f_rs, bn_gamma(base), bn_beta(base), total, N);
  };

  auto tblock = [&](float* x, const float* pos_p, const int* nb_p, int Np, int C, int base) {
    TBParams t = tbp(base);
    const int M = NBATCH * Np;
    float* h = f_tmp;
    gemm(x, t.w_in, t.b_in, h, M, C, C, 1);        // h = relu(x @ w_in + b_in)
    gemm(h, t.w_lin, nullptr, f_v,    M, C, C, 0); // v
    gemm(h, t.w_src, nullptr, f_asrc, M, C, C, 0); // alpha_src
    gemm(h, t.w_dst, nullptr, f_adst, M, C, C, 0); // alpha_dst
    size_t shmem = (size_t)(16 * C + 16 * PT_HID + C + 64) * sizeof(float) + 16 * sizeof(int);
    ptconv_kernel<<<dim3(M), dim3(128), shmem, stream>>>(
        pos_p, nb_p, f_v, f_asrc, f_adst,
        t.pw1, t.pb1, t.pw2, t.pb2, t.aw1, t.ab1, t.aw2, t.ab2,
        f_conv, Np, C);
    gemm(f_conv, t.w_out, t.b_out, x, M, C, C, 1); // x = relu(conv @ w_out + b_out)
  };

  // ---------------- stage 0: input MLP + first transformer block ----------------
  int Np = 1024, C = 128;
  bn_mlp(d_x, f_x, NBATCH * Np, 32, C, base_mlp_in);
  knn_kernel<<<dim3((NBATCH * Np + 255) / 256), dim3(256), 0, stream>>>(
      d_pos, d_pos, i_nb, Np, Np, 1, NBATCH * Np);
  tblock(f_x, d_pos, i_nb, Np, C, base_t_in);

  // ---------------- TransitionDown 1 + transformer block (C=256) ----------------
  const float* pos_cur = d_pos;
  {
    const int ns = Np / 2;  // 512
    fps_kernel<<<dim3(NBATCH), dim3(256), 0, stream>>>(pos_cur, i_fps, Np, ns);
    gather_pos_kernel<<<dim3((NBATCH * ns + 255) / 256), dim3(256), 0, stream>>>(
        pos_cur, i_fps, f_pos0, Np, ns, NBATCH * ns);
    bn_mlp(f_x, f_tmp, NBATCH * Np, C, 2 * C, base_tds0);   // xm [B,Np,2C]
    knn_kernel<<<dim3((NBATCH * ns + 255) / 256), dim3(256), 0, stream>>>(
        f_pos0, pos_cur, i_nbd, ns, Np, 0, NBATCH * ns);
    size_t total = (size_t)NBATCH * ns * 2 * C;
    maxpool_kernel<<<dim3((unsigned)((total + 255) / 256)), dim3(256), 0, stream>>>(
        f_tmp, i_nbd, f_x, Np, ns, 2 * C, total);
    pos_cur = f_pos0; Np = ns; C = 2 * C;                   // 512 pts, 256 ch
    knn_kernel<<<dim3((NBATCH * Np + 255) / 256), dim3(256), 0, stream>>>(
        pos_cur, pos_cur, i_nb, Np, Np, 1, NBATCH * Np);
    tblock(f_x, pos_cur, i_nb, Np, C, base_tfs0);
  }

  // ---------------- TransitionDown 2 + transformer block (C=512) ----------------
  {
    const int ns = Np / 2;  // 256
    fps_kernel<<<dim3(NBATCH), dim3(256), 0, stream>>>(pos_cur, i_fps, Np, ns);
    gather_pos_kernel<<<dim3((NBATCH * ns + 255) / 256), dim3(256), 0, stream>>>(
        pos_cur, i_fps, f_pos1, Np, ns, NBATCH * ns);
    bn_mlp(f_x, f_tmp, NBATCH * Np, C, 2 * C, base_tds1);   // xm [B,Np,2C]
    knn_kernel<<<dim3((NBATCH * ns + 255) / 256), dim3(256), 0, stream>>>(
        f_pos1, pos_cur, i_nbd, ns, Np, 0, NBATCH * ns);
    size_t total = (size_t)NBATCH * ns * 2 * C;
    maxpool_kernel<<<dim3((unsigned)((total + 255) / 256)), dim3(256), 0, stream>>>(
        f_tmp, i_nbd, f_x, Np, ns, 2 * C, total);
    pos_cur = f_pos1; Np = ns; C = 2 * C;                   // 256 pts, 512 ch
    knn_kernel<<<dim3((NBATCH * Np + 255) / 256), dim3(256), 0, stream>>>(
        pos_cur, pos_cur, i_nb, Np, Np, 1, NBATCH * Np);
    tblock(f_x, pos_cur, i_nb, Np, C, base_tfs1);
  }

  // ---------------- global mean pool + classification head ----------------
  meanpool_kernel<<<dim3(NBATCH), dim3(256), 0, stream>>>(f_x, f_pool, Np, C);
  gemm(f_pool, head_w, head_b, (float*)d_out, NBATCH, C, 256, 0);
}